// chunker_class_44951127720006
// MI455X (gfx1250) — compile-verified
//
#include <hip/hip_runtime.h>
#include <hip/hip_bf16.h>
#include <math.h>

// Problem constants (match reference)
#define Bdim 64
#define Sdim 512
#define Hdim 1024
#define Tdim 25
#define Npad 32                        // padded logit columns (2 WMMA N-tiles)
#define NTILES ((Bdim * Sdim) / 16)    // 2048 tiles of 16 pooled rows
#define WAVES_PER_BLOCK 4

typedef __attribute__((ext_vector_type(2))) float v2f;
typedef __attribute__((ext_vector_type(8))) float v8f;

// ---------------------------------------------------------------------------
// Kernel 0: zero-padded weight matrix Wpad[H][32] (cols 25..31 = 0) so the
// WMMA inner loop has no divergent/predicated B-fragment loads.
// ---------------------------------------------------------------------------
__global__ __launch_bounds__(256) void pad_w_kernel(const float* __restrict__ W,
                                                    float* __restrict__ Wpad) {
  const int idx = blockIdx.x * blockDim.x + threadIdx.x;  // 0 .. H*32-1
  if (idx >= Hdim * Npad) return;
  const int row = idx >> 5;
  const int col = idx & 31;
  Wpad[idx] = (col < Tdim) ? W[row * Tdim + col] : 0.0f;
}

// ---------------------------------------------------------------------------
// Kernel 1: per batch row, compute start index and token count of each
// contiguous segment slot k (segment_ids are nondecreasing per row).
// ---------------------------------------------------------------------------
__global__ __launch_bounds__(256) void seg_bounds_kernel(
    const int* __restrict__ seg, int* __restrict__ start, int* __restrict__ cnt) {
  __shared__ int s_cnt[Sdim];
  __shared__ int s_start[Sdim];
  const int b = blockIdx.x;
  for (int k = threadIdx.x; k < Sdim; k += blockDim.x) { s_cnt[k] = 0; s_start[k] = 0; }
  __syncthreads();
  const int* row = seg + (size_t)b * Sdim;
  for (int j = threadIdx.x; j < Sdim; j += blockDim.x) {
    int sg = row[j];
    atomicAdd(&s_cnt[sg], 1);
    if (j == 0 || row[j - 1] != sg) s_start[sg] = j;
  }
  __syncthreads();
  for (int k = threadIdx.x; k < Sdim; k += blockDim.x) {
    cnt[(size_t)b * Sdim + k] = s_cnt[k];
    start[(size_t)b * Sdim + k] = s_start[k];
  }
}

// ---------------------------------------------------------------------------
// Kernel 2: fused segment-mean-pool + GEMM(WMMA f32 16x16x4) + bias + ReLU +
// log-softmax NLL partials. One wave per 16-row pooled tile.
// ---------------------------------------------------------------------------
__global__ __launch_bounds__(32 * WAVES_PER_BLOCK) void fused_pool_gemm_nll_kernel(
    const float* __restrict__ hidden, const float* __restrict__ Wpad,
    const float* __restrict__ bias, const int* __restrict__ start,
    const int* __restrict__ cnt, const int* __restrict__ y,
    const int* __restrict__ amask, float* __restrict__ out_logits,
    float* __restrict__ tile_nll, float* __restrict__ tile_valid) {
  extern __shared__ float lds[];  // WAVES_PER_BLOCK * 16384 floats (64KB / wave)

  const int lane = threadIdx.x & 31;
  const int wave = threadIdx.x >> 5;
  const int tile = blockIdx.x * WAVES_PER_BLOCK + wave;
  const int b = tile >> 5;                 // 32 tiles per batch row (512/16)
  const int k0 = (tile & 31) << 4;
  const int rowBase = b * Sdim + k0;       // first pooled row of this tile
  float* At = lds + wave * 16384;          // A tile, transposed: At[h*16 + m]

  const int m = lane & 15;
  const int myStart = start[rowBase + m];
  const int myCnt = cnt[rowBase + m];
  const float myScale = (myCnt > 0) ? (1.0f / (float)myCnt) : 0.0f;

  // ---------------- Phase 1: build pooled A tile in LDS --------------------
  // Serialize over the 16 segment rows; wave-parallel over H with float4
  // coalesced streaming loads. Each hidden element is touched exactly once
  // across the whole grid.
  for (int mm = 0; mm < 16; ++mm) {
    const int st = __shfl(myStart, mm);
    const int cn = __shfl(myCnt, mm);
    const float sc = __shfl(myScale, mm);
    float4 acc[8];
#pragma unroll
    for (int i = 0; i < 8; ++i) acc[i] = make_float4(0.f, 0.f, 0.f, 0.f);
    const float* base = hidden + ((size_t)(b * Sdim + st)) * Hdim;
    for (int t = 0; t < cn; ++t) {
      const float4* src = (const float4*)(base + (size_t)t * Hdim);
      __builtin_prefetch(base + (size_t)(t + 2) * Hdim, 0, 1);
#pragma unroll
      for (int i = 0; i < 8; ++i) {
        float4 v = src[i * 32 + lane];
        acc[i].x += v.x; acc[i].y += v.y; acc[i].z += v.z; acc[i].w += v.w;
      }
    }
#pragma unroll
    for (int i = 0; i < 8; ++i) {
      const int h = i * 128 + lane * 4;
      At[(h + 0) * 16 + mm] = acc[i].x * sc;
      At[(h + 1) * 16 + mm] = acc[i].y * sc;
      At[(h + 2) * 16 + mm] = acc[i].z * sc;
      At[(h + 3) * 16 + mm] = acc[i].w * sc;
    }
  }
  // Same-wave LDS ops are in-order; compiler inserts s_wait_dscnt on use.

  // ---------------- Phase 2: WMMA K-loop (f32 16x16x4) ---------------------
  // A frag: lanes 0-15 -> M=0..15, VGPR0=K+0/K+2, VGPR1=K+1/K+3 (hi lanes +2).
  // B frags read from zero-padded Wpad: fully uniform control flow, EXEC
  // stays all-1s (required for WMMA), loads clause off one base address.
  v8f c0 = {};
  v8f c1 = {};
  const int kk = (lane >> 4) << 1;  // 0 for lanes 0-15, 2 for lanes 16-31
  const int n = lane & 15;
  const int col1 = 16 + n;
  for (int h0 = 0; h0 < Hdim; h0 += 4) {
    v2f a, b0f, b1f;
    a[0] = At[(h0 + kk) * 16 + m];
    a[1] = At[(h0 + kk + 1) * 16 + m];
    const float* wp = Wpad + (size_t)(h0 + kk) * Npad;
    b0f[0] = wp[n];
    b0f[1] = wp[Npad + n];
    b1f[0] = wp[n + 16];
    b1f[1] = wp[Npad + n + 16];
    c0 = __builtin_amdgcn_wmma_f32_16x16x4_f32(false, a, false, b0f, (short)0, c0,
                                               false, false);
    c1 = __builtin_amdgcn_wmma_f32_16x16x4_f32(false, a, false, b1f, (short)0, c1,
                                               false, false);
  }

  // ---------------- Epilogue: bias + ReLU, stage tile in LDS ---------------
  // D layout: VGPR v -> M = v (lanes 0-15) / v+8 (lanes 16-31); N = lane&15.
  // Padded columns (>=25) accumulated exact zeros, so relu(c1+0)=0 holds
  // unconditionally; only the bias load needs a bounds guard.
  float* lt = At;  // reuse wave-private LDS, row stride 33 to dodge banks
  const int mrow = (lane >> 4) * 8;
  const float bias0 = bias[n];
  const float bias1 = (col1 < Tdim) ? bias[col1] : 0.0f;
#pragma unroll
  for (int v = 0; v < 8; ++v) {
    float x0 = fmaxf(c0[v] + bias0, 0.0f);
    float x1 = fmaxf(c1[v] + bias1, 0.0f);
    lt[(mrow + v) * 33 + n] = x0;
    lt[(mrow + v) * 33 + col1] = x1;
  }

  // Coalesced logits store: this tile's output region is contiguous.
  for (int idx = lane; idx < 16 * Tdim; idx += 32) {
    const int rr = idx / Tdim;
    const int cc = idx - rr * Tdim;
    out_logits[(size_t)rowBase * Tdim + idx] = lt[rr * 33 + cc];
  }

  // Per-row log-softmax NLL (lanes 0-15 each own one row), then a
  // deterministic in-wave reduction to one partial per tile.
  float nll_l = 0.0f, val_l = 0.0f;
  if (lane < 16) {
    const int r = rowBase + lane;
    float mx = 0.0f;  // all values >= 0 after ReLU
    for (int tcol = 0; tcol < Tdim; ++tcol) mx = fmaxf(mx, lt[lane * 33 + tcol]);
    float s = 0.0f;
    for (int tcol = 0; tcol < Tdim; ++tcol) s += __expf(lt[lane * 33 + tcol] - mx);
    const float lse = mx + __logf(s);
    const int label = y[r];
    const float nll = lse - lt[lane * 33 + label];
    if (amask[r] == 1) { nll_l = nll; val_l = 1.0f; }
  }
#pragma unroll
  for (int off = 16; off > 0; off >>= 1) {
    nll_l += __shfl_down(nll_l, off);
    val_l += __shfl_down(val_l, off);
  }
  if (lane == 0) {
    tile_nll[tile] = nll_l;
    tile_valid[tile] = val_l;
  }
}

// ---------------------------------------------------------------------------
// Kernel 3: deterministic reduction of per-tile partials -> loss at d_out[0].
// ---------------------------------------------------------------------------
__global__ __launch_bounds__(256) void finalize_kernel(
    const float* __restrict__ tile_nll, const float* __restrict__ tile_valid,
    float* __restrict__ out) {
  __shared__ float s1[256];
  __shared__ float s2[256];
  float a = 0.0f, v = 0.0f;
  for (int i = threadIdx.x; i < NTILES; i += 256) {
    a += tile_nll[i];
    v += tile_valid[i];
  }
  s1[threadIdx.x] = a;
  s2[threadIdx.x] = v;
  __syncthreads();
  for (int off = 128; off > 0; off >>= 1) {
    if (threadIdx.x < off) {
      s1[threadIdx.x] += s1[threadIdx.x + off];
      s2[threadIdx.x] += s2[threadIdx.x + off];
    }
    __syncthreads();
  }
  if (threadIdx.x == 0) out[0] = s1[0] / fmaxf(s2[0], 1.0f);
}

// ---------------------------------------------------------------------------
extern "C" void kernel_launch(void* const* d_in, const int* in_sizes, int n_in,
                              void* d_out, int out_size, void* d_ws, size_t ws_size,
                              hipStream_t stream) {
  const float* hidden = (const float*)d_in[0];  // [B,S,H] f32
  const float* W      = (const float*)d_in[1];  // [H,T] f32
  const float* bias   = (const float*)d_in[2];  // [T] f32
  const int*   seg    = (const int*)d_in[3];    // [B,S] i32
  const int*   amask  = (const int*)d_in[4];    // [B,S] i32
  const int*   y      = (const int*)d_in[5];    // [B,S] i32
  float* out = (float*)d_out;                   // [1 + B*S*T] f32

  // Workspace layout (~410 KB): start/cnt tables, per-tile NLL partials,
  // zero-padded weights.
  int* start = (int*)d_ws;
  int* cnt = start + Bdim * Sdim;
  float* tile_nll = (float*)(cnt + Bdim * Sdim);
  float* tile_valid = tile_nll + NTILES;
  float* Wpad = tile_valid + NTILES;            // [H][32]

  pad_w_kernel<<<(Hdim * Npad + 255) / 256, 256, 0, stream>>>(W, Wpad);
  seg_bounds_kernel<<<Bdim, 256, 0, stream>>>(seg, start, cnt);

  const int blocks = NTILES / WAVES_PER_BLOCK;             // 512
  const size_t lds_bytes = WAVES_PER_BLOCK * 16384 * 4;    // 256 KB / block
  fused_pool_gemm_nll_kernel<<<blocks, 32 * WAVES_PER_BLOCK, lds_bytes, stream>>>(
      hidden, Wpad, bias, start, cnt, y, amask, out + 1, tile_nll, tile_valid);

  finalize_kernel<<<1, 256, 0, stream>>>(tile_nll, tile_valid, out);
}